// MaxUnpoolWithArgmax_9646496547553
// MI455X (gfx1250) — compile-verified
//
#include <hip/hip_runtime.h>

// MaxUnpoolWithArgmax  (B=16, H=64, W=64, C=128, 2x2 stride-2 windows)
//
// GATHER formulation: each thread owns one float4 (4 channels) of the pooled
// input, decodes the 4 argmax values (each lands in the thread's own 2x2
// output window), and writes all four window positions with coalesced
// non-temporal 128-bit stores (zeros everywhere except the winning slot).
// No zero-init pass, no atomics: 96 MiB read + 128 MiB write per call
// -> ~10 us at 23.3 TB/s HBM.

namespace {

// Native clang vectors (required by __builtin_nontemporal_*).
typedef float f32x4 __attribute__((ext_vector_type(4)));
typedef int   i32x4 __attribute__((ext_vector_type(4)));

constexpr int C_    = 128;   // channels
constexpr int WOUT_ = 128;   // W * 2
constexpr int HOUT_ = 128;   // H * 2

__global__ __launch_bounds__(256) void unpool_gather_kernel(
    const f32x4* __restrict__ in4,   // input viewed as float-x4   [nvec]
    const i32x4* __restrict__ am4,   // argmax int64 viewed as i32x4 (2 i64 / vec)
    float*       __restrict__ out,   // (16,128,128,128) f32
    int nvec)
{
    const int i = blockIdx.x * 256 + threadIdx.x;   // vec4 index into input
    if (i >= nvec) return;

    // Prefetch a future argmax cacheline into GL2 (global_prefetch_b8).
    {
        int j = i + 16384;
        if (j >= nvec) j = nvec - 1;
        __builtin_prefetch(am4 + 2 * (size_t)j, 0, 1);
    }

    // Streaming loads: nothing is reused, keep L2 clean (TH_NT).
    const f32x4 v  = __builtin_nontemporal_load(in4 + i);
    const i32x4 a0 = __builtin_nontemporal_load(am4 + 2 * (size_t)i);
    const i32x4 a1 = __builtin_nontemporal_load(am4 + 2 * (size_t)i + 1);

    // Decompose linear vec4 index (all dims are powers of two -> shifts).
    const int c  = (i & 31) << 2;        // channel (C/4 = 32 vec4 per (b,h,w))
    const int w  = (i >> 5)  & 63;
    const int h  = (i >> 11) & 63;
    const int b  =  i >> 17;
    const int y0 = h << 1;
    const int x0 = w << 1;

    // Decode argmax (max flat index < 2^21 -> low dword of each int64 suffices).
    //   y = idx >> 14 ; x = (idx >> 7) & 127 ; sel = (y - y0)*2 + (x - x0)
    const int idx0 = a0.x, idx1 = a0.z, idx2 = a1.x, idx3 = a1.z;
    const int s0 = (((idx0 >> 14) - y0) << 1) | (((idx0 >> 7) & 127) - x0);
    const int s1 = (((idx1 >> 14) - y0) << 1) | (((idx1 >> 7) & 127) - x0);
    const int s2 = (((idx2 >> 14) - y0) << 1) | (((idx2 >> 7) & 127) - x0);
    const int s3 = (((idx3 >> 14) - y0) << 1) | (((idx3 >> 7) & 127) - x0);

    // Build the four window-position vectors branchlessly (v_cndmask_b32).
    f32x4 o0, o1, o2, o3;
#define PICK(o, p)                         \
    o.x = (s0 == (p)) ? v.x : 0.0f;        \
    o.y = (s1 == (p)) ? v.y : 0.0f;        \
    o.z = (s2 == (p)) ? v.z : 0.0f;        \
    o.w = (s3 == (p)) ? v.w : 0.0f;
    PICK(o0, 0) PICK(o1, 1) PICK(o2, 2) PICK(o3, 3)
#undef PICK

    // Coalesced non-temporal 128-bit stores: one vec4 per window position.
    const size_t rowStride = (size_t)WOUT_ * C_;                    // 16384
    float* base = out + ((size_t)b * HOUT_ + (size_t)y0) * rowStride
                      + (size_t)x0 * C_ + (size_t)c;
    __builtin_nontemporal_store(o0, (f32x4*)(base));
    __builtin_nontemporal_store(o1, (f32x4*)(base + C_));
    __builtin_nontemporal_store(o2, (f32x4*)(base + rowStride));
    __builtin_nontemporal_store(o3, (f32x4*)(base + rowStride + C_));
}

} // namespace

extern "C" void kernel_launch(void* const* d_in, const int* in_sizes, int n_in,
                              void* d_out, int out_size, void* d_ws, size_t ws_size,
                              hipStream_t stream) {
    const f32x4* in4 = (const f32x4*)d_in[0];          // f32 inputs
    const i32x4* am4 = (const i32x4*)d_in[1];          // int64 argmax, viewed as i32x4
    float*       out = (float*)d_out;

    const int nvec   = in_sizes[0] / 4;                // 2,097,152 vec4s
    const int blocks = (nvec + 255) / 256;             // 8192 blocks of 8 wave32s

    unpool_gather_kernel<<<blocks, 256, 0, stream>>>(in4, am4, out, nvec);
}